// MedianFilterNonDifferentiable_52785148068080
// MI455X (gfx1250) — compile-verified
//
#include <hip/hip_runtime.h>
#include <stdint.h>

// ---------------------------------------------------------------------------
// 5x5 median filter, fp32, symmetric ("reflect") border.
// MI455X strategy:
//   * HBM traffic floor is ~100 MB (~4.3 us @ 23.3 TB/s); achieve read-once by
//     staging a 36x36 halo tile per 32x32 output tile into LDS.
//   * Use gfx1250 GLOBAL_LOAD_ASYNC_TO_LDS_B32 (ASYNCcnt) for the global->LDS
//     stage: no VGPR round trip, overlapped DMA-style copy.
//   * Median = order statistic -> pure min/max network in VGPRs (wave32 VOPD
//     dual-issues v_min/v_max). Forgetful selection: start with 14 of the 25
//     window values, repeatedly expel the working min and max (provably not
//     the median) and insert the next value; finish with median-of-3.
// ---------------------------------------------------------------------------

#define TILE   32
#define PAD    2
#define LW     (TILE + 2 * PAD)      // 36
#define LDSN   (LW * LW)             // 1296 floats = 5184 B LDS
#define NTHR   256                   // 8 wave32s
#define IMG_H  512
#define IMG_W  512

// Address-space-qualified pointer types for the gfx1250 async-LDS builtin
// (clang declares its params as addrspace(1)/addrspace(3) int*).
typedef __attribute__((address_space(1))) int g_as_int;
typedef __attribute__((address_space(3))) int l_as_int;

static __device__ __forceinline__ void s2(float& a, float& b) {
  const float lo = fminf(a, b);
  const float hi = fmaxf(a, b);
  a = lo;
  b = hi;
}

// Compare-exchange network: after the call, a[0] = min(a[0..N-1]) and
// a[N-1] = max(a[0..N-1]); the multiset of values is preserved.
// Cost ~ 3N/2 compare-exchanges.
template <int N>
static __device__ __forceinline__ void minmaxN(float (&a)[14]) {
  // Pair phase: even index <- pair min, odd index <- pair max.
#pragma unroll
  for (int i = 0; i + 1 < N; i += 2) s2(a[i], a[i + 1]);
  // Min-reduce pair-mins (even indices, incl. unpaired tail for odd N) into a[0].
#pragma unroll
  for (int i = 2; i < N; i += 2) s2(a[0], a[i]);
  // Max-reduce pair-maxes (odd indices) into a[N-1].
#pragma unroll
  for (int i = 1; i < N - 1; i += 2) s2(a[i], a[N - 1]);
}

// Window element J (row J/5, col J%5) relative to the window's top-left in LDS.
template <int J>
static __device__ __forceinline__ float wload(const float* wb) {
  return wb[(J / 5) * LW + (J % 5)];
}

// Forgetful selection over the remaining window elements. Working set size N.
// Invariant: #discarded_low == #discarded_high and N + #discarded_low == 14,
// so the working min (max) is <= (>=) 13 other elements and cannot be the
// median (rank 13 of 25). Inserts consume window elements 14..24.
template <int N>
static __device__ __forceinline__ float forget(float (&a)[14], const float* wb) {
  if constexpr (N == 3) {
    const float mn = fminf(a[0], a[1]);
    const float mx = fmaxf(a[0], a[1]);
    return fmaxf(mn, fminf(mx, a[2]));   // median of 3
  } else {
    minmaxN<N>(a);                       // a[0]=min, a[N-1]=max
    a[0] = wload<28 - N>(wb);            // drop min (overwrite) & max (shrink)
    return forget<N - 1>(a, wb);
  }
}

__global__ __launch_bounds__(NTHR) void median5x5_kernel(
    const float* __restrict__ in, float* __restrict__ out) {
  __shared__ float tile[LDSN];

  const int plane = blockIdx.z;                        // b*C + c, 0..47
  const size_t pbase = (size_t)plane * IMG_H * IMG_W;
  const float* src = in + pbase;
  float* dst = out + pbase;

  const int x0 = blockIdx.x * TILE;
  const int y0 = blockIdx.y * TILE;
  const int tid = threadIdx.x;

  // ---- Stage halo tile into LDS via async global->LDS loads --------------
#pragma unroll
  for (int it = 0; it < (LDSN + NTHR - 1) / NTHR; ++it) {
    const int idx = tid + it * NTHR;
    if (idx < LDSN) {
      const int ly = idx / LW;
      const int lx = idx % LW;
      int gy = y0 + ly - PAD;
      int gx = x0 + lx - PAD;
      // symmetric reflect: (d c b a | a b c d)
      gy = (gy < 0) ? (-gy - 1) : gy;
      gy = (gy >= IMG_H) ? (2 * IMG_H - 1 - gy) : gy;
      gx = (gx < 0) ? (-gx - 1) : gx;
      gx = (gx >= IMG_W) ? (2 * IMG_W - 1 - gx) : gx;

      const float* gptr = src + (size_t)gy * IMG_W + gx;
      float* lptr = &tile[idx];
#if __has_builtin(__builtin_amdgcn_global_load_async_to_lds_b32)
      __builtin_amdgcn_global_load_async_to_lds_b32(
          (g_as_int*)gptr, (l_as_int*)lptr, /*offset=*/0, /*cpol=*/0);
#else
      const uint32_t lds_addr =
          (uint32_t)(uintptr_t)(l_as_int*)lptr;
      const uint32_t goff = (uint32_t)(((size_t)gy * IMG_W + gx) * sizeof(float));
      asm volatile("global_load_async_to_lds_b32 %0, %1, %2"
                   :
                   : "v"(lds_addr), "v"(goff), "s"(src)
                   : "memory");
#endif
    }
  }

#if __has_builtin(__builtin_amdgcn_s_wait_asynccnt)
  __builtin_amdgcn_s_wait_asynccnt(0);
#else
  asm volatile("s_wait_asynccnt 0" ::: "memory");
#endif
  __syncthreads();

  // ---- Compute: 4 output pixels per thread -------------------------------
  const int lx = tid % TILE;       // output column within tile
  const int ty = tid / TILE;       // 0..7

#pragma unroll 1
  for (int k = 0; k < (TILE * TILE) / NTHR; ++k) {   // 4 iterations
    const int ly = ty + k * (NTHR / TILE);           // rows ty, ty+8, ty+16, ty+24
    const float* wb = &tile[ly * LW + lx];           // window top-left in LDS

    float a[14];
#pragma unroll
    for (int j = 0; j < 14; ++j) a[j] = wb[(j / 5) * LW + (j % 5)];

    const float med = forget<14>(a, wb);

    dst[(size_t)(y0 + ly) * IMG_W + (x0 + lx)] = med;
  }
}

extern "C" void kernel_launch(void* const* d_in, const int* in_sizes, int n_in,
                              void* d_out, int out_size, void* d_ws, size_t ws_size,
                              hipStream_t stream) {
  (void)n_in; (void)d_ws; (void)ws_size; (void)out_size;
  const float* img = (const float*)d_in[0];
  float* out = (float*)d_out;

  const int planes = in_sizes[0] / (IMG_H * IMG_W);   // 16*3 = 48
  dim3 grid(IMG_W / TILE, IMG_H / TILE, planes);      // 16 x 16 x 48
  dim3 block(NTHR, 1, 1);
  median5x5_kernel<<<grid, block, 0, stream>>>(img, out);
}